// DPSR_37890201485372
// MI455X (gfx1250) — compile-verified
//
#include <hip/hip_runtime.h>
#include <hip/hip_bf16.h>

// ---------------------------------------------------------------------------
// DPSR on gfx1250: rasterize -> 3D rFFT (DFT-as-WMMA-GEMM) -> spectral solve
// -> inverse 3D rFFT (WMMA) -> interp/normalize.
// All DFT passes use V_WMMA_F32_16X16X4_F32 against cos/sin twiddle tables.
// Passes are templated on input/output mode and on unit-stride so the inner
// K-loop is branch-free and vectorizes B loads where the data is contiguous.
// ---------------------------------------------------------------------------

namespace {
constexpr int   kRes   = 256;
constexpr int   kRH    = 129;              // rfft half-spectrum
constexpr int   kZP    = 144;              // padded to multiple of 16
constexpr long  kR3    = (long)kRes * kRes * kRes;   // 16777216
constexpr long  kPlane = (long)kRes * kRes * kZP;    // 9437184
constexpr long  kImOff = 3 * kPlane;                 // 28311552 (im block offset)
constexpr int   kNPts  = 500000;
constexpr float kSig   = 10.0f;
constexpr float kTwoPi = 6.28318530717958647692f;

// DFT pass modes
constexpr int kModeRealIn  = 0;  // Xre only -> complex out (forward z-rfft)
constexpr int kModeComplex = 1;  // complex -> complex (y/x forward+inverse)
constexpr int kModeRealOut = 2;  // complex -> real (inverse z-irfft)
}

typedef float v2f __attribute__((ext_vector_type(2)));
typedef float v8f __attribute__((ext_vector_type(8)));

// ---------------------------------------------------------------------------
// Utility: zero fill
// ---------------------------------------------------------------------------
__global__ void dpsr_zero(float* p, long n) {
  long i = (long)blockIdx.x * blockDim.x + threadIdx.x;
  long stride = (long)gridDim.x * blockDim.x;
  for (; i < n; i += stride) p[i] = 0.0f;
}

// ---------------------------------------------------------------------------
// Twiddle tables.
//  Ct[k*256+n] = cos(2pi k n/256), St = sin(...)        (forward & inverse via sign)
//  Crz[n*144+k] = w_k cos(2pi k n/256)/256  (k<=128, else 0)   w_0=w_128=1 else 2
//  Srz[n*144+k] = -w_k sin(2pi k n/256)/256
// Phase reduced mod 256 -> exact small-angle arguments.
// ---------------------------------------------------------------------------
__global__ void dpsr_tables(float* Ct, float* St, float* Crz, float* Srz) {
  int t = blockIdx.x * blockDim.x + threadIdx.x;
  if (t < kRes * kRes) {
    int k = t >> 8, n = t & 255;
    float a = (float)((k * n) & 255) * (kTwoPi / 256.0f);
    Ct[t] = cosf(a);
    St[t] = sinf(a);
  }
  if (t < kRes * kZP) {
    int n = t / kZP;
    int k = t - n * kZP;
    float cv = 0.0f, sv = 0.0f;
    if (k <= 128) {
      float wk = (k == 0 || k == 128) ? 1.0f : 2.0f;
      float a = (float)((k * n) & 255) * (kTwoPi / 256.0f);
      cv =  wk * cosf(a) * (1.0f / 256.0f);
      sv = -wk * sinf(a) * (1.0f / 256.0f);
    }
    Crz[t] = cv;
    Srz[t] = sv;
  }
}

// ---------------------------------------------------------------------------
// Trilinear corner setup (matches reference: ind0=floor, ind1=mod(ceil,256))
// ---------------------------------------------------------------------------
__device__ __forceinline__ void tri_setup(const float v[3], int i0[3], int i1[3],
                                          float f[3]) {
#pragma unroll
  for (int d = 0; d < 3; ++d) {
    float x = v[d] * 256.0f;
    float fl = floorf(x);
    i0[d] = ((int)fl) & 255;
    i1[d] = ((int)ceilf(x)) & 255;
    f[d] = x - fl;
  }
}

// ---------------------------------------------------------------------------
// Rasterize: scatter-add normals into 3 channel-major 256^3 grids.
// ---------------------------------------------------------------------------
__global__ void dpsr_rasterize(const float* __restrict__ V,
                               const float* __restrict__ Nrm,
                               float* __restrict__ ras) {
  int p = blockIdx.x * blockDim.x + threadIdx.x;
  if (p >= kNPts) return;
  float v[3] = {V[p * 3 + 0], V[p * 3 + 1], V[p * 3 + 2]};
  float nv[3] = {Nrm[p * 3 + 0], Nrm[p * 3 + 1], Nrm[p * 3 + 2]};
  int i0[3], i1[3];
  float f[3];
  tri_setup(v, i0, i1, f);
#pragma unroll
  for (int c = 0; c < 8; ++c) {
    int bx = (c >> 2) & 1, by = (c >> 1) & 1, bz = c & 1;
    int ix = bx ? i1[0] : i0[0];
    int iy = by ? i1[1] : i0[1];
    int iz = bz ? i1[2] : i0[2];
    float w = (bx ? f[0] : 1.0f - f[0]) * (by ? f[1] : 1.0f - f[1]) *
              (bz ? f[2] : 1.0f - f[2]);
    long idx = ((long)ix * kRes + iy) * kRes + iz;
#pragma unroll
    for (int ch = 0; ch < 3; ++ch)
      atomicAdd(&ras[(long)ch * kR3 + idx], w * nv[ch]);
  }
}

// ---------------------------------------------------------------------------
// Generic DFT pass as WMMA f32 GEMM.
//   Ore[k,l] = scale * ( sum_n Ct[k][n]*Xre[n,l]  + sS * sum_n St[k][n]*Xim[n,l] )
//   Oim[k,l] = scale * ( sum_n Ct[k][n]*Xim[n,l]  - sS * sum_n St[k][n]*Xre[n,l] )
// MODE kModeRealIn : Xim absent  -> Ore = C*re, Oim = -S*re
// MODE kModeComplex: full complex (4 accumulators)
// MODE kModeRealOut: Ore = C*re + sS*S*im only (2 accumulators)
// Lines addressed as  base = o*inOuter + i*inInner, element n at +n*strideN.
// One wave owns a 16x16 (freq x line) tile; EXEC always all-ones per wave.
// ---------------------------------------------------------------------------
struct DftArgs {
  const float* Xre;
  const float* Xim;
  const float* Ct;
  const float* St;
  float* Ore;
  float* Oim;
  int N;          // accumulation length (multiple of 4)
  int tabStride;  // twiddle row stride
  long inOuter, inInner, strideN;
  long outOuter, outInner, outK;
  int innerCount;  // multiple of 16
  float sS, scale;
  int numTiles, tilesM;
};

template <int MODE, bool UNIT>
__global__ __launch_bounds__(256) void dpsr_dft_wmma(DftArgs p) {
  const int wave = threadIdx.x >> 5;
  const int lane = threadIdx.x & 31;
  const int tile = blockIdx.x * 8 + wave;
  if (tile >= p.numTiles) return;  // wave-uniform exit

  const int tileM = tile % p.tilesM;
  const long lineBase = (long)(tile / p.tilesM) * 16;
  const long o  = lineBase / p.innerCount;
  const long i0 = lineBase - o * (long)p.innerCount;

  const int l15 = lane & 15;
  const int hi  = (lane >= 16) ? 1 : 0;

  // A operand: twiddle row for this lane; ISA layout: lanes>=16 hold K=2,3.
  const int arow = tileM * 16 + l15;
  const float* ctr = p.Ct + (long)arow * p.tabStride + (hi ? 2 : 0);
  const float* str = p.St + (long)arow * p.tabStride + (hi ? 2 : 0);

  // B operand: this lane supplies column (line) l15, K split mirroring A.
  const long sN = UNIT ? 1 : p.strideN;
  const long inBase =
      o * p.inOuter + (i0 + l15) * p.inInner + (long)(hi ? 2 : 0) * sN;
  const float* xre = p.Xre + inBase;
  const float* xim = (MODE != kModeRealIn) ? (p.Xim + inBase) : nullptr;

  v8f aCre = {}, aSre = {}, aCim = {}, aSim = {};
  for (int n0 = 0; n0 < p.N; n0 += 4) {
    v2f ac, as2;
    ac.x = ctr[n0];
    ac.y = ctr[n0 + 1];
    as2.x = str[n0];
    as2.y = str[n0 + 1];

    v2f bre, bim;
    if (UNIT) {
      if (MODE != kModeRealOut)
        bre = *reinterpret_cast<const v2f*>(xre + n0);   // 8B vector load
      else
        bre = *reinterpret_cast<const v2f*>(xre + n0);
      if (MODE != kModeRealIn)
        bim = *reinterpret_cast<const v2f*>(xim + n0);
    } else {
      __builtin_prefetch(&xre[(long)(n0 + 16) * sN], 0, 1);
      bre.x = xre[(long)n0 * sN];
      bre.y = xre[(long)(n0 + 1) * sN];
      if (MODE != kModeRealIn) {
        __builtin_prefetch(&xim[(long)(n0 + 16) * sN], 0, 1);
        bim.x = xim[(long)n0 * sN];
        bim.y = xim[(long)(n0 + 1) * sN];
      }
    }

    if (MODE == kModeRealIn) {
      aCre = __builtin_amdgcn_wmma_f32_16x16x4_f32(false, ac, false, bre,
                                                   (short)0, aCre, false, false);
      aSre = __builtin_amdgcn_wmma_f32_16x16x4_f32(false, as2, false, bre,
                                                   (short)0, aSre, false, false);
    } else if (MODE == kModeComplex) {
      aCre = __builtin_amdgcn_wmma_f32_16x16x4_f32(false, ac, false, bre,
                                                   (short)0, aCre, false, false);
      aSre = __builtin_amdgcn_wmma_f32_16x16x4_f32(false, as2, false, bre,
                                                   (short)0, aSre, false, false);
      aCim = __builtin_amdgcn_wmma_f32_16x16x4_f32(false, ac, false, bim,
                                                   (short)0, aCim, false, false);
      aSim = __builtin_amdgcn_wmma_f32_16x16x4_f32(false, as2, false, bim,
                                                   (short)0, aSim, false, false);
    } else {  // kModeRealOut: only C*re and S*im needed
      aCre = __builtin_amdgcn_wmma_f32_16x16x4_f32(false, ac, false, bre,
                                                   (short)0, aCre, false, false);
      aSim = __builtin_amdgcn_wmma_f32_16x16x4_f32(false, as2, false, bim,
                                                   (short)0, aSim, false, false);
    }
  }

  // C/D layout: VGPR r -> M = r (+8 for lanes 16..31), N = lane&15.
  const long outBase = o * p.outOuter + (i0 + l15) * p.outInner;
  const int mOff = tileM * 16 + (hi ? 8 : 0);
#pragma unroll
  for (int r = 0; r < 8; ++r) {
    long addr = outBase + (long)(mOff + r) * p.outK;
    if (MODE == kModeRealIn) {
      p.Ore[addr] = p.scale * aCre[r];
      p.Oim[addr] = p.scale * (-aSre[r]);
    } else if (MODE == kModeComplex) {
      p.Ore[addr] = p.scale * (aCre[r] + p.sS * aSim[r]);
      p.Oim[addr] = p.scale * (aCim[r] - p.sS * aSre[r]);
    } else {
      p.Ore[addr] = p.scale * (aCre[r] + p.sS * aSim[r]);
    }
  }
}

// ---------------------------------------------------------------------------
// Spectral solve:  Phi = (sum_d -i*Nhat_d*G*om_d) / (Lap + 1e-6), DC killed.
// Layout: (kx,ky,kz) flat over 256*256*144; kz>=129 zeroed.
// ---------------------------------------------------------------------------
__global__ void dpsr_spectral(const float* __restrict__ nh,
                              float* __restrict__ phre,
                              float* __restrict__ phim) {
  long t = (long)blockIdx.x * blockDim.x + threadIdx.x;
  if (t >= (long)kRes * kRes * kZP) return;
  int kz = (int)(t % kZP);
  long r = t / kZP;
  int ky = (int)(r & 255);
  int kx = (int)(r >> 8);
  if (kz >= kRH || (kx == 0 && ky == 0 && kz == 0)) {
    phre[t] = 0.0f;
    phim[t] = 0.0f;
    return;
  }
  float fx = (kx < 128) ? (float)kx : (float)(kx - 256);
  float fy = (ky < 128) ? (float)ky : (float)(ky - 256);
  float fz = (float)kz;
  float d2 = fx * fx + fy * fy + fz * fz;
  float s = kSig * 2.0f / 256.0f;
  float g = expf(-0.5f * s * s * d2);
  float ox = kTwoPi * fx, oy = kTwoPi * fy, oz = kTwoPi * fz;
  float lap = -(ox * ox + oy * oy + oz * oz);
  float inv = 1.0f / (lap + 1e-6f);
  float om[3] = {ox, oy, oz};
  float dre = 0.0f, dim = 0.0f;
#pragma unroll
  for (int ch = 0; ch < 3; ++ch) {
    float nr = nh[(long)ch * kPlane + t];
    float ni = nh[kImOff + (long)ch * kPlane + t];
    float go = g * om[ch];
    dre += ni * go;   // (-i)(a+ib) = b - ia
    dim -= nr * go;
  }
  phre[t] = dre * inv;
  phim[t] = dim * inv;
}

// ---------------------------------------------------------------------------
// Interpolate phi at points, block-reduce, atomic sum into scal[0].
// ---------------------------------------------------------------------------
__global__ void dpsr_interp_reduce(const float* __restrict__ V,
                                   const float* __restrict__ phi,
                                   float* __restrict__ scal) {
  __shared__ float red[256];
  int p = blockIdx.x * blockDim.x + threadIdx.x;
  float acc = 0.0f;
  if (p < kNPts) {
    float v[3] = {V[p * 3 + 0], V[p * 3 + 1], V[p * 3 + 2]};
    int i0[3], i1[3];
    float f[3];
    tri_setup(v, i0, i1, f);
#pragma unroll
    for (int c = 0; c < 8; ++c) {
      int bx = (c >> 2) & 1, by = (c >> 1) & 1, bz = c & 1;
      int ix = bx ? i1[0] : i0[0];
      int iy = by ? i1[1] : i0[1];
      int iz = bz ? i1[2] : i0[2];
      float w = (bx ? f[0] : 1.0f - f[0]) * (by ? f[1] : 1.0f - f[1]) *
                (bz ? f[2] : 1.0f - f[2]);
      acc += w * phi[((long)ix * kRes + iy) * kRes + iz];
    }
  }
  red[threadIdx.x] = acc;
  __syncthreads();
  for (int s = 128; s > 0; s >>= 1) {
    if (threadIdx.x < s) red[threadIdx.x] += red[threadIdx.x + s];
    __syncthreads();
  }
  if (threadIdx.x == 0) atomicAdd(&scal[0], red[0]);
}

// mean & scale into scratch (avoids read-elem0 / write race in normalize).
__global__ void dpsr_finalize_prep(const float* __restrict__ phi,
                                   float* __restrict__ scal) {
  if (threadIdx.x == 0 && blockIdx.x == 0) {
    float mean = scal[0] * (1.0f / (float)kNPts);
    float fv0 = phi[0] - mean;
    scal[1] = mean;
    scal[2] = -0.5f / fabsf(fv0);
  }
}

__global__ void dpsr_normalize(float* __restrict__ phi,
                               const float* __restrict__ scal) {
  long i = (long)blockIdx.x * blockDim.x + threadIdx.x;
  if (i < kR3) phi[i] = (phi[i] - scal[1]) * scal[2];
}

// ---------------------------------------------------------------------------
// Host orchestration
// ---------------------------------------------------------------------------
template <int MODE, bool UNIT>
static void launch_dft(const DftArgs& a, hipStream_t s) {
  hipLaunchKernelGGL((dpsr_dft_wmma<MODE, UNIT>), dim3(a.numTiles / 8),
                     dim3(256), 0, s, a);
}

extern "C" void kernel_launch(void* const* d_in, const int* in_sizes, int n_in,
                              void* d_out, int out_size, void* d_ws,
                              size_t ws_size, hipStream_t stream) {
  (void)in_sizes; (void)n_in; (void)out_size; (void)ws_size;
  const float* V = (const float*)d_in[0];
  const float* N = (const float*)d_in[1];
  float* phi = (float*)d_out;

  float* w    = (float*)d_ws;
  float* ras  = w;                          // 3 * 16777216
  float* bufA = ras + 3 * kR3;              // 2*3*kPlane (re block | im block)
  float* bufB = bufA + 2 * kImOff;
  float* Ct   = bufB + 2 * kImOff;          // 65536
  float* St   = Ct + kRes * kRes;           // 65536
  float* Crz  = St + kRes * kRes;           // 36864
  float* Srz  = Crz + kRes * kZP;           // 36864
  float* scal = Srz + kRes * kZP;           // 4

  // 0) zero accumulators
  hipLaunchKernelGGL(dpsr_zero, dim3(16384), dim3(256), 0, stream, ras,
                     (long)3 * kR3);
  hipLaunchKernelGGL(dpsr_zero, dim3(1), dim3(256), 0, stream, scal, (long)4);

  // 1) twiddle tables
  hipLaunchKernelGGL(dpsr_tables, dim3(256), dim3(256), 0, stream, Ct, St, Crz,
                     Srz);

  // 2) rasterize
  hipLaunchKernelGGL(dpsr_rasterize, dim3((kNPts + 255) / 256), dim3(256), 0,
                     stream, V, N, ras);

  DftArgs a;
  a.Ct = Ct; a.St = St; a.tabStride = kRes;

  // 3) forward rfft along z: real (ch,x,y,z) -> complex (ch,x,y,k<144)
  a.Xre = ras; a.Xim = nullptr;
  a.Ore = bufA; a.Oim = bufA + kImOff;
  a.N = 256;
  a.inOuter = 65536; a.inInner = 256; a.strideN = 1;
  a.outOuter = 36864; a.outInner = kZP; a.outK = 1;
  a.innerCount = 256; a.sS = 1.0f; a.scale = 1.0f;
  a.tilesM = kZP / 16;
  a.numTiles = a.tilesM * (768 * 256 / 16);   // 110592
  launch_dft<kModeRealIn, true>(a, stream);

  // 4) forward fft along y
  a.Xre = bufA; a.Xim = bufA + kImOff;
  a.Ore = bufB; a.Oim = bufB + kImOff;
  a.inOuter = 36864; a.inInner = 1; a.strideN = kZP;
  a.outOuter = 36864; a.outInner = 1; a.outK = kZP;
  a.innerCount = kZP;
  a.tilesM = 16;
  a.numTiles = 16 * (768 * kZP / 16);         // 110592
  launch_dft<kModeComplex, false>(a, stream);

  // 5) forward fft along x -> Nhat in bufA
  a.Xre = bufB; a.Xim = bufB + kImOff;
  a.Ore = bufA; a.Oim = bufA + kImOff;
  a.inOuter = kPlane; a.inInner = 1; a.strideN = 36864;
  a.outOuter = kPlane; a.outInner = 1; a.outK = 36864;
  a.innerCount = 36864;
  a.numTiles = 16 * (3 * 36864 / 16);         // 110592
  launch_dft<kModeComplex, false>(a, stream);

  // 6) spectral solve -> Phi (re in bufB[0], im in bufB[kImOff])
  hipLaunchKernelGGL(dpsr_spectral,
                     dim3((unsigned)((kRes * (long)kRes * kZP) / 256)),
                     dim3(256), 0, stream, bufA, bufB, bufB + kImOff);

  // 7) inverse fft along x
  a.Xre = bufB; a.Xim = bufB + kImOff;
  a.Ore = bufA; a.Oim = bufA + kImOff;
  a.inOuter = kPlane; a.inInner = 1; a.strideN = 36864;
  a.outOuter = kPlane; a.outInner = 1; a.outK = 36864;
  a.innerCount = 36864;
  a.sS = -1.0f; a.scale = 1.0f / 256.0f;
  a.numTiles = 16 * (36864 / 16);             // 36864
  launch_dft<kModeComplex, false>(a, stream);

  // 8) inverse fft along y
  a.Xre = bufA; a.Xim = bufA + kImOff;
  a.Ore = bufB; a.Oim = bufB + kImOff;
  a.inOuter = 36864; a.inInner = 1; a.strideN = kZP;
  a.outOuter = 36864; a.outInner = 1; a.outK = kZP;
  a.innerCount = kZP;
  a.numTiles = 16 * (256 * kZP / 16);         // 36864
  launch_dft<kModeComplex, false>(a, stream);

  // 9) inverse rfft along z -> real phi in d_out (weights baked in Crz/Srz)
  a.Xre = bufB; a.Xim = bufB + kImOff;
  a.Ct = Crz; a.St = Srz; a.tabStride = kZP;
  a.Ore = phi; a.Oim = nullptr;
  a.N = kZP;
  a.inOuter = 36864; a.inInner = kZP; a.strideN = 1;
  a.outOuter = 65536; a.outInner = 256; a.outK = 1;
  a.innerCount = 256;
  a.sS = 1.0f; a.scale = 1.0f;
  a.numTiles = 16 * (65536 / 16);             // 65536
  launch_dft<kModeRealOut, true>(a, stream);

  // 10) interpolate at points, mean-shift, normalize
  hipLaunchKernelGGL(dpsr_interp_reduce, dim3((kNPts + 255) / 256), dim3(256),
                     0, stream, V, phi, scal);
  hipLaunchKernelGGL(dpsr_finalize_prep, dim3(1), dim3(64), 0, stream, phi,
                     scal);
  hipLaunchKernelGGL(dpsr_normalize, dim3((unsigned)(kR3 / 256)), dim3(256), 0,
                     stream, phi, scal);
}